// Encoder_9740985827606
// MI455X (gfx1250) — compile-verified
//
#include <hip/hip_runtime.h>
#include <cstddef>

typedef float v2f __attribute__((ext_vector_type(2)));
typedef float v8f __attribute__((ext_vector_type(8)));

// ---------------------------------------------------------------------------
// FP32 atomic add -> native global_atomic_add_f32 (relaxed, device scope)
// ---------------------------------------------------------------------------
__device__ __forceinline__ void atomic_add_f32(float* p, float v) {
  __hip_atomic_fetch_add(p, v, __ATOMIC_RELAXED, __HIP_MEMORY_SCOPE_AGENT);
}

// ---------------------------------------------------------------------------
// WMMA fp32 GEMM: C[M,Nc] = A[M,K] * B[K,Nc] (+ bias[Nc]) (+ ReLU)
// Each wave computes a 16(M) x 64(N) strip using 4 v_wmma_f32_16x16x4_f32
// accumulators; 4 waves per block handle 4 consecutive M-tiles.
// Requires: M % 16 == 0, K % 4 == 0, Nc % 64 == 0.
// ---------------------------------------------------------------------------
#define GEMM_WAVES 4

__global__ void __launch_bounds__(GEMM_WAVES * 32)
gemm_wmma_f32(const float* __restrict__ A, const float* __restrict__ B,
              float* __restrict__ C, const float* __restrict__ bias,
              int M, int Nc, int K, int relu) {
  const int wave = threadIdx.x >> 5;
  const int lane = threadIdx.x & 31;
  const int mtile = blockIdx.x * GEMM_WAVES + wave;
  if (mtile * 16 >= M) return;           // wave-uniform: EXEC stays all-ones

  const int m0 = mtile * 16;
  const int n0 = blockIdx.y * 64;
  const int lm = lane & 15;              // column within 16-wide tile
  const int hi = lane >> 4;              // 0: lanes 0-15, 1: lanes 16-31
  const int lk = hi * 2;                 // K sub-offset for A/B fragments

  v8f acc[4];
#pragma unroll
  for (int t = 0; t < 4; ++t) acc[t] = (v8f){};

  const float* arow = A + (size_t)(m0 + lm) * K;

  for (int k = 0; k < K; k += 4) {
    // A fragment: lane holds A[m0+lm][k+lk], A[m0+lm][k+lk+1]
    v2f a = *(const v2f*)(arow + k + lk);

    // B fragments for the 4 N-tiles
    const float* bp = B + (size_t)(k + lk) * Nc + n0 + lm;
    v2f bf[4];
#pragma unroll
    for (int t = 0; t < 4; ++t) {
      bf[t].x = bp[16 * t];              // row k+lk
      bf[t].y = bp[(size_t)Nc + 16 * t]; // row k+lk+1
    }

#pragma unroll
    for (int t = 0; t < 4; ++t) {
      acc[t] = __builtin_amdgcn_wmma_f32_16x16x4_f32(
          /*neg_a=*/false, a, /*neg_b=*/false, bf[t],
          /*c_mod=*/(short)0, acc[t], /*reuse_a=*/false, /*reuse_b=*/false);
    }
  }

  // Epilogue: VGPR j -> row m0 + j + 8*hi, col n0 + 16*t + lm
#pragma unroll
  for (int t = 0; t < 4; ++t) {
    const int col = n0 + 16 * t + lm;
    const float bv = bias ? bias[col] : 0.0f;
#pragma unroll
    for (int j = 0; j < 8; ++j) {
      const int r = m0 + j + hi * 8;
      float v = acc[t][j] + bv;
      if (relu) v = fmaxf(v, 0.0f);
      C[(size_t)r * Nc + col] = v;
    }
  }
}

// ---------------------------------------------------------------------------
// Degree / normalization helpers
// ---------------------------------------------------------------------------
__global__ void fill_ones(float* __restrict__ p, int n) {
  int i = blockIdx.x * blockDim.x + threadIdx.x;
  if (i < n) p[i] = 1.0f;                // self-loop contributes 1 to degree
}

__global__ void deg_count(float* __restrict__ deg, const int* __restrict__ row,
                          int E) {
  int i = blockIdx.x * blockDim.x + threadIdx.x;
  if (i < E) atomic_add_f32(&deg[row[i]], 1.0f);
}

__global__ void to_rsqrt(float* __restrict__ p, int n) {
  int i = blockIdx.x * blockDim.x + threadIdx.x;
  if (i < n) p[i] = rsqrtf(p[i]);        // deg >= 1 always
}

// aggr[i,f] = dinv[i]^2 * xw[i,f]  (self-loop message, also initializes aggr)
__global__ void selfloop_init(float* __restrict__ aggr,
                              const float* __restrict__ xw,
                              const float* __restrict__ dinv, int n, int F) {
  int i = blockIdx.x * blockDim.x + threadIdx.x;
  int total = n * F;
  if (i < total) {
    int node = i / F;
    float d = dinv[node];
    aggr[i] = d * d * xw[i];
  }
}

// One wave per edge: aggr[row] += dinv[row]*dinv[col] * xw[col]
__global__ void edge_scatter(float* __restrict__ aggr,
                             const float* __restrict__ xw,
                             const float* __restrict__ dinv,
                             const int* __restrict__ row,
                             const int* __restrict__ col, int E, int F) {
  long long gid = (long long)blockIdx.x * blockDim.x + threadIdx.x;
  int e = (int)(gid >> 5);
  int lane = (int)(gid & 31);
  if (e >= E) return;
  const int r = row[e];
  const int c = col[e];
  const float norm = dinv[r] * dinv[c];
  const float* __restrict__ src = xw + (size_t)c * F;
  float* __restrict__ dst = aggr + (size_t)r * F;
  for (int f = lane; f < F; f += 32) {
    atomic_add_f32(dst + f, norm * src[f]);
  }
}

// z[i,f] = (z[i,f] + b[f]) (optionally ReLU); F must be a power of two
__global__ void bias_act(float* __restrict__ z, const float* __restrict__ b,
                         int n, int F, int relu) {
  int i = blockIdx.x * blockDim.x + threadIdx.x;
  int total = n * F;
  if (i < total) {
    float v = z[i] + b[i & (F - 1)];
    if (relu) v = fmaxf(v, 0.0f);
    z[i] = v;
  }
}

// ---------------------------------------------------------------------------
// Orchestration
// ---------------------------------------------------------------------------
extern "C" void kernel_launch(void* const* d_in, const int* in_sizes, int n_in,
                              void* d_out, int out_size, void* d_ws,
                              size_t ws_size, hipStream_t stream) {
  const float* x   = (const float*)d_in[0];
  const int*   ei  = (const int*)d_in[1];
  const float* W1  = (const float*)d_in[2];
  const float* b1  = (const float*)d_in[3];
  const float* W2  = (const float*)d_in[4];
  const float* b2  = (const float*)d_in[5];
  const float* Wmu = (const float*)d_in[6];
  const float* bmu = (const float*)d_in[7];
  const float* Wlv = (const float*)d_in[8];
  const float* blv = (const float*)d_in[9];

  const int Nn = in_sizes[0] / 256;   // 100000 nodes
  const int E  = in_sizes[1] / 2;     // 3.2M edges
  const int* row = ei;
  const int* col = ei + E;

  float* ws   = (float*)d_ws;
  float* dinv = ws;                                   // Nn floats
  float* bufA = ws + (((size_t)Nn + 1023) & ~(size_t)1023);  // Nn*256
  float* bufB = bufA + (size_t)Nn * 256;              // Nn*256
  float* out  = (float*)d_out;                        // mu | logvar

  const int EB = 256;
  // --- degrees: deg = 1 (self loop) + count over row; dinv = rsqrt(deg) ---
  fill_ones<<<(Nn + EB - 1) / EB, EB, 0, stream>>>(dinv, Nn);
  deg_count<<<(E + EB - 1) / EB, EB, 0, stream>>>(dinv, row, E);
  to_rsqrt<<<(Nn + EB - 1) / EB, EB, 0, stream>>>(dinv, Nn);

  const int mtiles = Nn / 16;                         // 6250
  const int gx = (mtiles + GEMM_WAVES - 1) / GEMM_WAVES;
  const long long scatterThreads = (long long)E * 32;
  const int scatterBlocks = (int)((scatterThreads + EB - 1) / EB);

  // --- layer 1: xw1 = x @ W1 ; aggregate ; + b1 ; ReLU ---
  gemm_wmma_f32<<<dim3(gx, 256 / 64), GEMM_WAVES * 32, 0, stream>>>(
      x, W1, bufA, nullptr, Nn, 256, 256, 0);
  selfloop_init<<<(Nn * 256 + EB - 1) / EB, EB, 0, stream>>>(bufB, bufA, dinv,
                                                             Nn, 256);
  edge_scatter<<<scatterBlocks, EB, 0, stream>>>(bufB, bufA, dinv, row, col, E,
                                                 256);
  bias_act<<<(Nn * 256 + EB - 1) / EB, EB, 0, stream>>>(bufB, b1, Nn, 256, 1);

  // --- layer 2: xw2 = z1 @ W2 ; aggregate ; + b2 ---
  float* xw2   = bufA;                   // Nn*128 (reuses xw1 space)
  float* aggr2 = bufA + (size_t)Nn * 128;
  gemm_wmma_f32<<<dim3(gx, 128 / 64), GEMM_WAVES * 32, 0, stream>>>(
      bufB, W2, xw2, nullptr, Nn, 128, 256, 0);
  selfloop_init<<<(Nn * 128 + EB - 1) / EB, EB, 0, stream>>>(aggr2, xw2, dinv,
                                                             Nn, 128);
  edge_scatter<<<scatterBlocks, EB, 0, stream>>>(aggr2, xw2, dinv, row, col, E,
                                                 128);
  bias_act<<<(Nn * 128 + EB - 1) / EB, EB, 0, stream>>>(aggr2, b2, Nn, 128, 0);

  // --- heads: mu = z2 @ Wmu + bmu ; logvar = z2 @ Wlv + blv ---
  gemm_wmma_f32<<<dim3(gx, 128 / 64), GEMM_WAVES * 32, 0, stream>>>(
      aggr2, Wmu, out, bmu, Nn, 128, 128, 0);
  gemm_wmma_f32<<<dim3(gx, 128 / 64), GEMM_WAVES * 32, 0, stream>>>(
      aggr2, Wlv, out + (size_t)Nn * 128, blv, Nn, 128, 128, 0);
}